// SpatialGate_32753420599525
// MI455X (gfx1250) — compile-verified
//
#include <hip/hip_runtime.h>
#include <cstdint>
#include <cstddef>

typedef float v4f __attribute__((ext_vector_type(4)));

// Problem constants (fixed by the reference).
constexpr int B   = 16;
constexpr int C   = 128;
constexpr int H   = 224;
constexpr int W   = 224;
constexpr int HW  = H * W;        // 50176
constexpr int HW4 = HW / 4;       // 12544
constexpr int CHW = C * HW;       // 6422528
constexpr int BHW = B * HW;       // 802816

// Issue one async global->LDS b128 copy (CDNA5 data mover).
//  - s_wait_dscnt 0: drain the ds_load of this buffer from two stages ago
//    before the async engine may overwrite it (WAR on LDS).
//  - s_wait_asynccnt 1 after issuing: async loads complete in order per
//    wave, so <=1 outstanding guarantees the *previous* stage has landed.
//  - th:TH_LOAD_NT: the x stream is dead after one read; keep L2 for the
//    pooled planes.
#define ASYNC_STAGE(ldsdst, gaddr)                                   \
  asm volatile("s_wait_dscnt 0\n\t"                                  \
               "global_load_async_to_lds_b128 %0, %1, off th:TH_LOAD_NT\n\t" \
               "s_wait_asynccnt 1"                                   \
               :: "v"(ldsdst), "v"(gaddr)                            \
               : "memory")

__device__ __forceinline__ void reduce4(v4f& vmax, v4f& vsum, const v4f v) {
  vmax.x = fmaxf(vmax.x, v.x);
  vmax.y = fmaxf(vmax.y, v.y);
  vmax.z = fmaxf(vmax.z, v.z);
  vmax.w = fmaxf(vmax.w, v.w);
  vsum += v;
}

// ---------------------------------------------------------------------------
// Kernel 1: channel max + mean pooling.
// One float4 of pixels per thread; 128-channel loop as a 2-stage async
// global->LDS double buffer, unrolled 2x so buffer parity is static
// (constant ds offsets, no per-iteration branch), tail peeled.
// Grid exact: BHW/4 = 200704 = 784 * 256 threads.
// ---------------------------------------------------------------------------
__global__ void __launch_bounds__(256) pool_kernel(
    const float* __restrict__ x,
    float* __restrict__ omax,
    float* __restrict__ omean) {
  __shared__ v4f buf[2][256];

  const int t   = threadIdx.x;
  const int g   = blockIdx.x * 256 + t;     // float4-pixel index over B*HW/4
  const int b   = g / HW4;
  const int hw4 = g - b * HW4;

  const uint64_t gbase =
      (uint64_t)(uintptr_t)(x + (size_t)b * CHW + (size_t)hw4 * 4);
  const uint64_t cstride = (uint64_t)HW * 4u;   // bytes between channels
  const unsigned lds0 = (unsigned)(uintptr_t)(&buf[0][t]);
  const unsigned lds1 = (unsigned)(uintptr_t)(&buf[1][t]);

  // Prime the pipeline: channel 0 -> buf0.
  asm volatile("global_load_async_to_lds_b128 %0, %1, off th:TH_LOAD_NT"
               :: "v"(lds0), "v"(gbase)
               : "memory");

  v4f vmax = {-__builtin_inff(), -__builtin_inff(),
              -__builtin_inff(), -__builtin_inff()};
  v4f vsum = {0.0f, 0.0f, 0.0f, 0.0f};

  // Steady state: channels 0..125 consumed, 1..126 prefetched.
  for (int c = 0; c < C - 2; c += 2) {
    ASYNC_STAGE(lds1, gbase + (uint64_t)(c + 1) * cstride);
    reduce4(vmax, vsum, buf[0][t]);           // channel c   (buf0 ready)
    ASYNC_STAGE(lds0, gbase + (uint64_t)(c + 2) * cstride);
    reduce4(vmax, vsum, buf[1][t]);           // channel c+1 (buf1 ready)
  }
  // Tail: c = 126.
  ASYNC_STAGE(lds1, gbase + (uint64_t)(C - 1) * cstride);
  reduce4(vmax, vsum, buf[0][t]);             // channel 126
  asm volatile("s_wait_asynccnt 0" ::: "memory");
  reduce4(vmax, vsum, buf[1][t]);             // channel 127

  const size_t o = (size_t)b * HW + (size_t)hw4 * 4;
  *(v4f*)(omax + o)  = vmax;
  *(v4f*)(omean + o) = vsum * (1.0f / 128.0f);
}

// ---------------------------------------------------------------------------
// Kernel 2: 3x3 conv (2->1 ch, zero pad 1) over (max, mean) planes + sigmoid.
// One pixel per thread. Inputs are L2-hot (6.4 MB). Grid exact: 3136 * 256.
// ---------------------------------------------------------------------------
__global__ void __launch_bounds__(256) conv_kernel(
    const float* __restrict__ pmax,
    const float* __restrict__ pmean,
    const float* __restrict__ cw,   // [1,2,3,3]: [0..8]=max-plane, [9..17]=mean-plane
    const float* __restrict__ cb,   // [1]
    float* __restrict__ scale) {
  const int g  = blockIdx.x * 256 + threadIdx.x;  // pixel index over B*HW
  const int b  = g / HW;
  const int hw = g - b * HW;
  const int h  = hw / W;
  const int w  = hw - h * W;

  float wm[9], wn[9];
#pragma unroll
  for (int k = 0; k < 9; ++k) {   // uniform -> scalar loads
    wm[k] = cw[k];
    wn[k] = cw[9 + k];
  }

  float y = cb[0];
#pragma unroll
  for (int kh = 0; kh < 3; ++kh) {
    const int hh = h + kh - 1;
    if (hh < 0 || hh >= H) continue;
#pragma unroll
    for (int kw = 0; kw < 3; ++kw) {
      const int ww = w + kw - 1;
      if (ww < 0 || ww >= W) continue;
      const size_t q = (size_t)b * HW + (size_t)hh * W + ww;
      y = fmaf(wm[kh * 3 + kw], pmax[q], y);
      y = fmaf(wn[kh * 3 + kw], pmean[q], y);
    }
  }
  scale[g] = 1.0f / (1.0f + __expf(-y));
}

// ---------------------------------------------------------------------------
// Kernel 3: out = x * scale (broadcast over channels).
// NT load x / NT store out (411 MB each, zero reuse -> don't thrash L2);
// scale stays RT (3.2 MB, reused 128x across channels, L2-resident).
// Grid exact: B*C*HW/4 = 25690112 = 100352 * 256.
// ---------------------------------------------------------------------------
__global__ void __launch_bounds__(256) mul_kernel(
    const float* __restrict__ x,
    const float* __restrict__ scale,
    float* __restrict__ out) {
  const int i4  = blockIdx.x * 256 + threadIdx.x;   // float4 index
  const int b   = i4 / (C * HW4);
  const int r   = i4 - b * (C * HW4);
  const int hw4 = r % HW4;

  const v4f xv = __builtin_nontemporal_load((const v4f*)x + (size_t)i4);
  const v4f sv = *((const v4f*)scale + (size_t)b * HW4 + hw4);
  __builtin_nontemporal_store(xv * sv, (v4f*)out + (size_t)i4);
}

// ---------------------------------------------------------------------------
extern "C" void kernel_launch(void* const* d_in, const int* in_sizes, int n_in,
                              void* d_out, int out_size, void* d_ws, size_t ws_size,
                              hipStream_t stream) {
  (void)in_sizes; (void)n_in; (void)out_size; (void)ws_size;

  const float* x  = (const float*)d_in[0];
  const float* cw = (const float*)d_in[1];
  const float* cb = (const float*)d_in[2];
  float* out = (float*)d_out;

  // Workspace layout: [max BHW][mean BHW][scale BHW] = 9.6 MB total.
  float* wmax   = (float*)d_ws;
  float* wmean  = wmax + BHW;
  float* wscale = wmean + BHW;

  pool_kernel<<<(BHW / 4) / 256, 256, 0, stream>>>(x, wmax, wmean);
  conv_kernel<<<BHW / 256, 256, 0, stream>>>(wmax, wmean, cw, cb, wscale);
  mul_kernel<<<((size_t)B * C * HW / 4) / 256, 256, 0, stream>>>(x, wscale, out);
}